// ModuleFlowProj_41583873359893
// MI455X (gfx1250) — compile-verified
//
#include <hip/hip_runtime.h>
#include <stdint.h>

#define BB 4
#define HH 1024
#define WW 1920
#define HW (HH * WW)
#define NN (BB * HW)

#define RB 256          // rowfill block size
#define EPT 8           // elements per thread in row scan (8*256 >= 1920)

// Splat tiling: 64x48 interior, 8-px halo -> 80x64 accumulation region.
#define TX 64
#define TY 48
#define RAD 8
#define AX (TX + 2 * RAD)      // 80
#define AY (TY + 2 * RAD)      // 64
#define ACELLS (AX * AY)       // 5120 cells; 3 arrays = 61440 B LDS

// ---------------------------------------------------------------------------
// Zero the accumulator workspace (acc_x, acc_y, cnt are contiguous).
// ---------------------------------------------------------------------------
__global__ void fp_zero_kernel(float* __restrict__ p, int n) {
    int i = blockIdx.x * blockDim.x + threadIdx.x;
    int stride = gridDim.x * blockDim.x;
    for (; i < n; i += stride) p[i] = 0.0f;
}

// ---------------------------------------------------------------------------
// Splat with LDS privatization: accumulate (-fx,-fy,1) for the 4 integer
// neighbors of (x+fx, y+fy) into an LDS tile (ds_add_f32), then flush the
// tile once with global_atomic_add_f32. Splats escaping the 8-px halo
// (~8-sigma for N(0,1) flow) fall back to direct global atomics.
// Out-of-image taps are dropped by the bounds check at flush time.
// ---------------------------------------------------------------------------
__global__ __launch_bounds__(256) void splat_kernel(
        const float* __restrict__ in,
        float* __restrict__ accx, float* __restrict__ accy,
        float* __restrict__ cnt) {
    __shared__ float s_ax[ACELLS];
    __shared__ float s_ay[ACELLS];
    __shared__ float s_c [ACELLS];

    const int tid = threadIdx.x;
    const int b   = blockIdx.z;
    const int tileX0 = blockIdx.x * TX;
    const int tileY0 = blockIdx.y * TY;

    // Zero LDS accumulators.
    for (int i = tid; i < ACELLS; i += 256) {
        s_ax[i] = 0.0f; s_ay[i] = 0.0f; s_c[i] = 0.0f;
    }
    __syncthreads();

    // 256 threads cover the 64x48 interior: tx = tid%64, ty strides by 4.
    const int tx = tid & 63;
    const int ty0 = tid >> 6;          // 0..3
    const int inbase = b * 2 * HW;

    for (int k = 0; k < TY / 4; ++k) {
        int ty = ty0 + 4 * k;
        int gx = tileX0 + tx;          // always < WW (1920 = 30*64)
        int gy = tileY0 + ty;
        if (gy >= HH) continue;        // partial bottom tiles (1024 = 21*48+16)

        float fx = in[inbase + gy * WW + gx];
        float fy = in[inbase + HW + gy * WW + gx];
        int xf = (int)floorf((float)gx + fx);
        int yf = (int)floorf((float)gy + fy);
        float nfx = -fx, nfy = -fy;

        int lx = xf - (tileX0 - RAD);
        int ly = yf - (tileY0 - RAD);
        if (lx >= 0 && lx + 1 < AX && ly >= 0 && ly + 1 < AY) {
            // Fast path: whole 2x2 footprint inside the LDS tile.
#pragma unroll
            for (int dy = 0; dy < 2; ++dy) {
#pragma unroll
                for (int dx = 0; dx < 2; ++dx) {
                    int c = (ly + dy) * AX + (lx + dx);
                    atomicAdd(&s_ax[c], nfx);   // ds_add_f32
                    atomicAdd(&s_ay[c], nfy);
                    atomicAdd(&s_c [c], 1.0f);
                }
            }
        } else {
            // Rare escape: direct bounds-checked global atomics.
#pragma unroll
            for (int dy = 0; dy < 2; ++dy) {
#pragma unroll
                for (int dx = 0; dx < 2; ++dx) {
                    int xi = xf + dx, yi = yf + dy;
                    if (xi >= 0 && xi < WW && yi >= 0 && yi < HH) {
                        int o = b * HW + yi * WW + xi;
                        unsafeAtomicAdd(&accx[o], nfx);
                        unsafeAtomicAdd(&accy[o], nfy);
                        unsafeAtomicAdd(&cnt[o], 1.0f);
                    }
                }
            }
        }
    }
    __syncthreads();

    // Flush tile: only touched cells that map inside the image.
    for (int c = tid; c < ACELLS; c += 256) {
        float cc = s_c[c];
        if (cc != 0.0f) {
            int lx = c % AX;
            int ly = c / AX;
            int gx = tileX0 - RAD + lx;
            int gy = tileY0 - RAD + ly;
            if (gx >= 0 && gx < WW && gy >= 0 && gy < HH) {
                int o = b * HW + gy * WW + gx;
                unsafeAtomicAdd(&accx[o], s_ax[c]);  // global_atomic_add_f32
                unsafeAtomicAdd(&accy[o], s_ay[c]);
                unsafeAtomicAdd(&cnt[o], cc);
            }
        }
    }
}

// ---------------------------------------------------------------------------
// Row fill: one workgroup per row.
//  - Stage acc_x/acc_y/cnt row slices into LDS with CDNA5 async global->LDS
//    copies (ASYNCcnt), then normalize in place.
//  - Hillis-Steele scans in LDS: nearest-valid-left (max-scan of index) and
//    nearest-valid-right (min-scan).
//  - Write averaged value for valid pixels, left/right fill for invalid ones.
//  - Flag rows that contain no valid pixel (handled by column pass).
// ---------------------------------------------------------------------------
__global__ __launch_bounds__(RB) void rowfill_kernel(
        const float* __restrict__ accx, const float* __restrict__ accy,
        const float* __restrict__ cnt, float* __restrict__ out,
        int* __restrict__ rowEmpty) {
    __shared__ float s_vx[WW];
    __shared__ float s_vy[WW];
    __shared__ float s_c[WW];
    __shared__ int   s_nl[WW];
    __shared__ int   s_nr[WW];

    const int r = blockIdx.x;          // 0 .. B*H-1
    const int b = r / HH;
    const int y = r - b * HH;
    const int rowbase = b * HW + y * WW;
    const int tid = threadIdx.x;

    unsigned long long pax = (unsigned long long)(uintptr_t)(accx + rowbase);
    unsigned long long pay = (unsigned long long)(uintptr_t)(accy + rowbase);
    unsigned long long pc  = (unsigned long long)(uintptr_t)(cnt  + rowbase);
    // Low 32 bits of a generic pointer to __shared__ = LDS byte offset
    unsigned lvx0 = (unsigned)(uintptr_t)(&s_vx[0]);
    unsigned lvy0 = (unsigned)(uintptr_t)(&s_vy[0]);
    unsigned lc0  = (unsigned)(uintptr_t)(&s_c[0]);

    for (int x = tid; x < WW; x += RB) {
        unsigned off = (unsigned)(x * 4);
        asm volatile("global_load_async_to_lds_b32 %0, %1, %2"
                     :: "v"(lvx0 + off), "v"(off), "s"(pax) : "memory");
        asm volatile("global_load_async_to_lds_b32 %0, %1, %2"
                     :: "v"(lvy0 + off), "v"(off), "s"(pay) : "memory");
        asm volatile("global_load_async_to_lds_b32 %0, %1, %2"
                     :: "v"(lc0 + off), "v"(off), "s"(pc) : "memory");
    }
    asm volatile("s_wait_asynccnt 0" ::: "memory");
    __syncthreads();

    // Normalize in place; seed scan arrays.
    for (int x = tid; x < WW; x += RB) {
        float c = s_c[x];
        bool valid = (c > 0.0f);
        float inv = valid ? (1.0f / c) : 0.0f;  // invalid => acc is exactly 0
        s_vx[x] *= inv;
        s_vy[x] *= inv;
        s_nl[x] = valid ? x : -1;
        s_nr[x] = valid ? x : WW;
    }
    __syncthreads();

    // Hillis-Steele: max-scan (nearest valid <= x), min-scan (nearest valid >= x)
    for (int off = 1; off < WW; off <<= 1) {
        int rl[EPT], rr[EPT];
#pragma unroll
        for (int e = 0; e < EPT; ++e) {
            int x = tid + e * RB;
            if (x < WW) {
                int a = s_nl[x];
                if (x >= off) { int q = s_nl[x - off]; a = (a > q) ? a : q; }
                rl[e] = a;
                int c = s_nr[x];
                if (x + off < WW) { int q = s_nr[x + off]; c = (c < q) ? c : q; }
                rr[e] = c;
            }
        }
        __syncthreads();
#pragma unroll
        for (int e = 0; e < EPT; ++e) {
            int x = tid + e * RB;
            if (x < WW) { s_nl[x] = rl[e]; s_nr[x] = rr[e]; }
        }
        __syncthreads();
    }

    // Emit: own value if valid, else nearest-left, else nearest-right.
    const int obase = b * 2 * HW + y * WW;
    for (int x = tid; x < WW; x += RB) {
        int nl = s_nl[x];
        int nr = s_nr[x];
        int src = (nl >= 0) ? nl : ((nr < WW) ? nr : -1);
        if (src >= 0) {
            out[obase + x]      = s_vx[src];
            out[obase + HW + x] = s_vy[src];
        }
    }
    if (tid == 0) rowEmpty[r] = (s_nl[WW - 1] < 0) ? 1 : 0;
}

// ---------------------------------------------------------------------------
// Column fill: only rows with zero valid pixels remain. One thread per column
// (fully coalesced across x). Pass 1 top->bottom carries the nearest valid
// value above; pixels above the first valid row take the value AT the first
// valid row (nearest-below == firstValid), or 0 if the column is empty.
// ---------------------------------------------------------------------------
__global__ void colfill_kernel(const float* __restrict__ cnt,
                               const int* __restrict__ rowEmpty,
                               float* __restrict__ out) {
    int x = blockIdx.x * blockDim.x + threadIdx.x;
    int b = blockIdx.y;
    if (x >= WW) return;
    const int base  = b * HW;
    const int obase = b * 2 * HW;

    bool found = false;
    float cx = 0.0f, cy = 0.0f;
    float dvx = 0.0f, dvy = 0.0f;
    int firstValid = HH;

    for (int y = 0; y < HH; ++y) {
        int idx = base + y * WW + x;
        if (y + 8 < HH) __builtin_prefetch(cnt + idx + 8 * WW, 0, 0);  // global_prefetch_b8
        float c = cnt[idx];
        if (c > 0.0f) {
            cx = out[obase + y * WW + x];
            cy = out[obase + HW + y * WW + x];
            if (!found) { firstValid = y; dvx = cx; dvy = cy; }
            found = true;
        } else if (rowEmpty[b * HH + y] && found) {
            out[obase + y * WW + x]      = cx;
            out[obase + HW + y * WW + x] = cy;
        }
    }
    // Above the first valid row: nearest-below value (or 0 if column empty).
    for (int y = 0; y < firstValid; ++y) {
        if (rowEmpty[b * HH + y]) {
            out[obase + y * WW + x]      = dvx;
            out[obase + HW + y * WW + x] = dvy;
        }
    }
}

// ---------------------------------------------------------------------------
extern "C" void kernel_launch(void* const* d_in, const int* in_sizes, int n_in,
                              void* d_out, int out_size, void* d_ws, size_t ws_size,
                              hipStream_t stream) {
    const float* in = (const float*)d_in[0];
    float* out = (float*)d_out;

    float* ws   = (float*)d_ws;
    float* accx = ws;                   // NN floats
    float* accy = ws + (size_t)NN;      // NN floats
    float* cntv = ws + 2 * (size_t)NN;  // NN floats
    int* rowEmpty = (int*)(ws + 3 * (size_t)NN);  // B*H ints

    fp_zero_kernel<<<2048, 256, 0, stream>>>(accx, 3 * NN);
    splat_kernel<<<dim3(WW / TX, (HH + TY - 1) / TY, BB), 256, 0, stream>>>(
        in, accx, accy, cntv);
    rowfill_kernel<<<BB * HH, RB, 0, stream>>>(accx, accy, cntv, out, rowEmpty);
    colfill_kernel<<<dim3((WW + 255) / 256, BB), 256, 0, stream>>>(cntv, rowEmpty, out);
}